// ReconstructionLoss_WeightedMAE_PatchSSIM_88691074662843
// MI455X (gfx1250) — compile-verified
//
#include <hip/hip_runtime.h>
#include <math.h>

// CDNA5 f32 WMMA: D(16x16) = A(16x4) * B(4x16) + C
typedef __attribute__((ext_vector_type(2))) float v2f;
typedef __attribute__((ext_vector_type(4))) float v4f;
typedef __attribute__((ext_vector_type(8))) float v8f;

#define WMMA_F32_16x16x4(a, b, c) \
  __builtin_amdgcn_wmma_f32_16x16x4_f32(false, (a), false, (b), (short)0, (c), false, false)

namespace {
constexpr int   kTPB       = 256;         // 8 waves (wave32)
constexpr int   kTiles     = 4096;        // B*nh*nw*C = 8*8*8*8 patch-channel tiles
constexpr float kC1        = 1.0e-4f;     // (0.01 * 1.0)^2
constexpr float kC2        = 9.0e-4f;     // (0.03 * 1.0)^2
constexpr float kSsimCount = 11943936.0f; // 4096 tiles * 54 * 54
}

// One block per 64x64 patch-channel tile.
// Fused: MAE partials + patch SSIM via banded-matmul separable Gaussian conv on WMMA.
__global__ __launch_bounds__(kTPB) void fused_loss_tile_kernel(
    const float* __restrict__ inp, const float* __restrict__ outp,
    const float* __restrict__ tgt, float* __restrict__ partial)
{
  __shared__ float s_recon[4096];    // recon tile, row-major [r][c]
  __shared__ float s_tgt[4096];      // target tile, row-major [r][c]
  __shared__ float s_GT[4096];       // GT[j][k] = g[k-j] (0 if j>=54 or k-j not in [0,11))
  __shared__ float s_h[5][4096];     // H-pass results, transposed: XhT[j][r] per signal

  const int p  = blockIdx.x;
  const int c  = p & 7;
  const int pw = (p >> 3) & 7;
  const int ph = (p >> 6) & 7;
  const int b  = p >> 9;
  const int base = (((b << 3) + c) * 512 + (ph << 6)) * 512 + (pw << 6);
  const int t = threadIdx.x;

  // Normalized 1D Gaussian (WIN=11, SIGMA=1.5): g[i] = exp(-(i-5)^2/4.5) / sum
  float gsum = 0.f;
  #pragma unroll
  for (int i = 0; i < 11; ++i) {
    float d = (float)i - 5.f;
    gsum += expf(-d * d * (1.0f / 4.5f));
  }
  const float ginv = 1.f / gsum;

  // ---- Phase 0: stream tiles (non-temporal b128), build recon/target in LDS, MAE ----
  float macc = 0.f, wacc = 0.f;
  #pragma unroll
  for (int s0 = 0; s0 < 4; ++s0) {
    const int q   = t + kTPB * s0;      // float4 index within tile (0..1023)
    const int row = q >> 4;
    const int c4  = (q & 15) << 2;
    const int g   = base + row * 512 + c4;
    const v4f vi = __builtin_nontemporal_load(reinterpret_cast<const v4f*>(inp + g));
    const v4f vo = __builtin_nontemporal_load(reinterpret_cast<const v4f*>(outp + g));
    const v4f vt = __builtin_nontemporal_load(reinterpret_cast<const v4f*>(tgt + g));
    const v4f vr = vi + vo;
    *reinterpret_cast<v4f*>(&s_recon[row * 64 + c4]) = vr;
    *reinterpret_cast<v4f*>(&s_tgt[row * 64 + c4])   = vt;
    #pragma unroll
    for (int e = 0; e < 4; ++e) {
      const float hp = vt[e] - vi[e];
      const float w  = fabsf(hp) > 0.f ? 1.f : 0.f;
      macc += w * fabsf(vo[e] - hp);
      wacc += w;
    }
  }

  // Banded Gaussian matrix (transposed): GT[j][k], zero rows j>=54 keep padding exact.
  #pragma unroll
  for (int n = 0; n < 16; ++n) {
    const int e = t + kTPB * n;
    const int j = e >> 6, k = e & 63;
    const int d = k - j;
    float val = 0.f;
    if (j < 54 && d >= 0 && d < 11) {
      float dd = (float)d - 5.f;
      val = expf(-dd * dd * (1.0f / 4.5f)) * ginv;
    }
    s_GT[e] = val;
  }
  __syncthreads();

  const int lane = t & 31;
  // Wave index as an explicit scalar: tile/k loop bounds become wave-uniform, so the
  // compiler emits scalar (SCC) loop control and EXEC stays all-1s across every WMMA,
  // as the ISA requires (avoids the divergent exec-masked loop pattern).
  const int wv = __builtin_amdgcn_readfirstlane(t >> 5);
  const int lh = lane & 15;   // A: M-lane / B: N-lane
  const int hh = lane >> 4;   // K-half selector (K pair = k0+2*hh, +1)

  // ---- H-pass: Xh = X * Gh  (5 signals share A loads and B) ----
  for (int tl = wv; tl < 16; tl += 8) {
    const int M0 = (tl >> 2) << 4;
    const int j0 = (tl & 3) << 4;
    const int nsteps = (j0 == 48) ? 4 : 7;   // band: K in [j0, j0+25], clamp K<=63
    v8f a0 = {}, a1 = {}, a2 = {}, a3 = {}, a4 = {};
    for (int kk = 0; kk < nsteps; ++kk) {
      const int col = j0 + 4 * kk + 2 * hh;
      const v2f rA = *reinterpret_cast<const v2f*>(&s_recon[(M0 + lh) * 64 + col]);
      const v2f tA = *reinterpret_cast<const v2f*>(&s_tgt[(M0 + lh) * 64 + col]);
      const v2f B  = *reinterpret_cast<const v2f*>(&s_GT[(j0 + lh) * 64 + col]);
      a0 = WMMA_F32_16x16x4(rA,      B, a0);   // conv(r)
      a1 = WMMA_F32_16x16x4(tA,      B, a1);   // conv(t)
      a2 = WMMA_F32_16x16x4(rA * rA, B, a2);   // conv(r*r)
      a3 = WMMA_F32_16x16x4(tA * tA, B, a3);   // conv(t*t)
      a4 = WMMA_F32_16x16x4(rA * tA, B, a4);   // conv(r*t)
    }
    const int jrow = (j0 + lh) * 64 + M0 + 8 * hh;  // store transposed XhT[j][r]
    #pragma unroll
    for (int v = 0; v < 8; ++v) {
      s_h[0][jrow + v] = a0[v];
      s_h[1][jrow + v] = a1[v];
      s_h[2][jrow + v] = a2[v];
      s_h[3][jrow + v] = a3[v];
      s_h[4][jrow + v] = a4[v];
    }
  }
  __syncthreads();

  // ---- V-pass: Out = Gv * Xh ; Gv[i][k] = GT[i][k]. Then SSIM map + local sum ----
  float ssum = 0.f;
  for (int tl = wv; tl < 16; tl += 8) {
    const int i0 = (tl >> 2) << 4;
    const int j0 = (tl & 3) << 4;
    const int nsteps = (i0 == 48) ? 4 : 7;   // band: K in [i0, i0+25], clamp K<=63
    v8f m1 = {}, m2 = {}, crr = {}, ctt = {}, crt = {};
    for (int kk = 0; kk < nsteps; ++kk) {
      const int col  = i0 + 4 * kk + 2 * hh;
      const v2f A    = *reinterpret_cast<const v2f*>(&s_GT[(i0 + lh) * 64 + col]);
      const int boff = (j0 + lh) * 64 + col;
      const v2f b0 = *reinterpret_cast<const v2f*>(&s_h[0][boff]);
      const v2f b1 = *reinterpret_cast<const v2f*>(&s_h[1][boff]);
      const v2f b2 = *reinterpret_cast<const v2f*>(&s_h[2][boff]);
      const v2f b3 = *reinterpret_cast<const v2f*>(&s_h[3][boff]);
      const v2f b4 = *reinterpret_cast<const v2f*>(&s_h[4][boff]);
      m1  = WMMA_F32_16x16x4(A, b0, m1);
      m2  = WMMA_F32_16x16x4(A, b1, m2);
      crr = WMMA_F32_16x16x4(A, b2, crr);
      ctt = WMMA_F32_16x16x4(A, b3, ctt);
      crt = WMMA_F32_16x16x4(A, b4, crt);
    }
    #pragma unroll
    for (int v = 0; v < 8; ++v) {
      const int oi = i0 + v + 8 * hh;
      const int oj = j0 + lh;
      if (oi < 54 && oj < 54) {
        const float mu1 = m1[v], mu2 = m2[v];
        const float mu1s = mu1 * mu1, mu2s = mu2 * mu2, mu12 = mu1 * mu2;
        const float s1  = crr[v] - mu1s;
        const float s2  = ctt[v] - mu2s;
        const float s12 = crt[v] - mu12;
        const float num = (2.f * mu12 + kC1) * (2.f * s12 + kC2);
        const float den = (mu1s + mu2s + kC1) * (s1 + s2 + kC2);
        ssum += num / den;
      }
    }
  }

  // ---- Deterministic block reduction (reuse s_recon; unused after H-pass) ----
  s_recon[t]       = ssum;
  s_recon[256 + t] = macc;
  s_recon[512 + t] = wacc;
  __syncthreads();
  for (int st = 128; st > 0; st >>= 1) {
    if (t < st) {
      s_recon[t]       += s_recon[t + st];
      s_recon[256 + t] += s_recon[256 + t + st];
      s_recon[512 + t] += s_recon[512 + t + st];
    }
    __syncthreads();
  }
  if (t == 0) {
    partial[p * 4 + 0] = s_recon[0];
    partial[p * 4 + 1] = s_recon[256];
    partial[p * 4 + 2] = s_recon[512];
  }
}

__global__ __launch_bounds__(kTPB) void finalize_loss_kernel(
    const float* __restrict__ partial, float* __restrict__ out)
{
  __shared__ float red[768];
  const int t = threadIdx.x;
  float s = 0.f, m = 0.f, w = 0.f;
  for (int i = t; i < kTiles; i += kTPB) {
    s += partial[i * 4 + 0];
    m += partial[i * 4 + 1];
    w += partial[i * 4 + 2];
  }
  red[t] = s; red[256 + t] = m; red[512 + t] = w;
  __syncthreads();
  for (int st = 128; st > 0; st >>= 1) {
    if (t < st) {
      red[t]       += red[t + st];
      red[256 + t] += red[256 + t + st];
      red[512 + t] += red[512 + t + st];
    }
    __syncthreads();
  }
  if (t == 0) {
    const float wmae      = red[256] / (red[512] + 1e-8f);
    const float ssim_mean = red[0] / kSsimCount;
    out[0] = 0.5f * wmae + 0.5f * (1.0f - ssim_mean);
  }
}

extern "C" void kernel_launch(void* const* d_in, const int* in_sizes, int n_in,
                              void* d_out, int out_size, void* d_ws, size_t ws_size,
                              hipStream_t stream) {
  (void)in_sizes; (void)n_in; (void)out_size; (void)ws_size;
  const float* inp  = (const float*)d_in[0];   // input_image
  const float* outp = (const float*)d_in[1];   // output_image
  const float* tgt  = (const float*)d_in[2];   // target_image
  float* partial = (float*)d_ws;               // 4096 * 4 floats = 64 KB
  float* o = (float*)d_out;

  fused_loss_tile_kernel<<<dim3(kTiles), dim3(kTPB), 0, stream>>>(inp, outp, tgt, partial);
  finalize_loss_kernel<<<dim3(1), dim3(kTPB), 0, stream>>>(partial, o);
}